// GCN3DClassifier_25847113187817
// MI455X (gfx1250) — compile-verified
//
#include <hip/hip_runtime.h>
#include <hip/hip_bf16.h>
#include <cstddef>

typedef __attribute__((ext_vector_type(2))) float v2f;
typedef __attribute__((ext_vector_type(8))) float v8f;

#if __has_builtin(__builtin_amdgcn_global_load_async_to_lds_b128) && \
    __has_builtin(__builtin_amdgcn_s_wait_asynccnt)
#define HAS_ASYNC_LDS 1
#else
#define HAS_ASYNC_LDS 0
#endif

namespace {
constexpr int kNodes   = 100000;
constexpr int kEdges   = 1600000;
constexpr int kGraphs  = 64;
constexpr int kInDim   = 64;
constexpr int kHid     = 128;
constexpr int kHalfHid = 64;
constexpr int kNcls    = 10;
constexpr int kLdsStride = 132;   // 128 + 4 pad: breaks 64-bank conflict on row reads
constexpr int kScanBlks  = (kNodes + 255) / 256;   // 391
}

__device__ __forceinline__ void atomAddF(float* p, float v) {
  unsafeAtomicAdd(p, v);          // native global_atomic_add_f32
}

#if HAS_ASYNC_LDS
// Async 16B global->LDS copy (GLOBAL_LOAD_ASYNC_TO_LDS_B128, ASYNCcnt-tracked).
// Builtin signature (from clang diagnostic): (AS1 v4i* src, AS3 v4i* dst, imm offset, imm cpol)
typedef int v4i __attribute__((vector_size(16)));
typedef __attribute__((address_space(1))) v4i* gv4i_p;
typedef __attribute__((address_space(3))) v4i* sv4i_p;
__device__ __forceinline__ void asyncCopy16(const float* g, float* l) {
  __builtin_amdgcn_global_load_async_to_lds_b128(
      (gv4i_p)const_cast<float*>(g), (sv4i_p)l, /*offset=*/0, /*cpol=*/0);
}
#endif

// ---------------- generic init ----------------
__global__ void k_zero_f(float* __restrict__ p, int n) {
  int i = blockIdx.x * blockDim.x + threadIdx.x;
  if (i < n) p[i] = 0.f;
}
__global__ void k_zero_i(int* __restrict__ p, int n) {
  int i = blockIdx.x * blockDim.x + threadIdx.x;
  if (i < n) p[i] = 0;
}

// ---------------- degree count (int atomics) + dinv ----------------
__global__ void k_count(const int* __restrict__ dst, int* __restrict__ cnt) {
  int e = blockIdx.x * blockDim.x + threadIdx.x;
  if (e < kEdges) atomicAdd(&cnt[dst[e]], 1);
}
__global__ void k_dinv(const int* __restrict__ cnt, float* __restrict__ dinv) {
  int i = blockIdx.x * blockDim.x + threadIdx.x;
  if (i < kNodes) dinv[i] = rsqrtf((float)cnt[i] + 1.0f);   // +1 self-loop
}

// ---------------- 3-phase exclusive scan: cnt -> rowptr ----------------
__global__ __launch_bounds__(256) void k_scan_p1(const int* __restrict__ cnt,
                                                 int* __restrict__ rowptr,
                                                 int* __restrict__ partial) {
  __shared__ int s[256];
  int i = blockIdx.x * 256 + threadIdx.x;
  int v = (i < kNodes) ? cnt[i] : 0;
  s[threadIdx.x] = v;
  __syncthreads();
  for (int off = 1; off < 256; off <<= 1) {
    int t = (threadIdx.x >= off) ? s[threadIdx.x - off] : 0;
    __syncthreads();
    s[threadIdx.x] += t;
    __syncthreads();
  }
  if (i < kNodes) rowptr[i] = s[threadIdx.x] - v;           // block-local exclusive
  if (threadIdx.x == 255) partial[blockIdx.x] = s[255];     // block total
}
__global__ __launch_bounds__(512) void k_scan_p2(int* __restrict__ partial) {
  __shared__ int s[512];
  int v = (threadIdx.x < kScanBlks) ? partial[threadIdx.x] : 0;
  s[threadIdx.x] = v;
  __syncthreads();
  for (int off = 1; off < 512; off <<= 1) {
    int t = (threadIdx.x >= off) ? s[threadIdx.x - off] : 0;
    __syncthreads();
    s[threadIdx.x] += t;
    __syncthreads();
  }
  if (threadIdx.x < kScanBlks) partial[threadIdx.x] = s[threadIdx.x] - v;
}
__global__ __launch_bounds__(256) void k_scan_p3(int* __restrict__ rowptr,
                                                 const int* __restrict__ partial) {
  int i = blockIdx.x * 256 + threadIdx.x;
  if (i < kNodes) rowptr[i] += partial[blockIdx.x];
  if (i == 0) rowptr[kNodes] = kEdges;                      // total is known statically
}

// ---------------- CSR fill: adj[] holds src of each incoming edge ----------------
__global__ void k_fill_adj(const int* __restrict__ src, const int* __restrict__ dst,
                           int* __restrict__ rowwork, int* __restrict__ adj) {
  int e = blockIdx.x * blockDim.x + threadIdx.x;
  if (e >= kEdges) return;
  int pos = atomicAdd(&rowwork[dst[e]], 1);
  adj[pos] = src[e];
}

// ---------------- WMMA f32 GEMM: C[M,128] = A[M,K] @ W[K,128] ----------------
// Block = 256 threads = 8 waves. Block owns 16 rows; wave w owns cols [16w,16w+16).
// M multiple of 16 (100000 = 6250*16) -> no divergence, EXEC all-ones for WMMA.
// A tile staged to LDS with async global->LDS B128 copies when available.
__global__ __launch_bounds__(256) void k_gemm_wmma(
    const float* __restrict__ A, const float* __restrict__ W,
    float* __restrict__ C, int K) {
  __shared__ float sA[16 * kLdsStride];
  const int row0 = blockIdx.x << 4;

  const int kf4 = K >> 2;
  for (int i = threadIdx.x; i < 16 * kf4; i += 256) {
    int r  = i / kf4;
    int c4 = (i - r * kf4) << 2;
    const float* g = A + (size_t)(row0 + r) * K + c4;
    float* l = &sA[r * kLdsStride + c4];
#if HAS_ASYNC_LDS
    asyncCopy16(g, l);                              // bypasses VGPRs, ASYNCcnt
#else
    *reinterpret_cast<float4*>(l) = *reinterpret_cast<const float4*>(g);
#endif
  }
#if HAS_ASYNC_LDS
  __builtin_amdgcn_s_wait_asynccnt(0);              // drain this wave's async copies
#endif
  __syncthreads();                                  // publish LDS across waves

  const int lane = threadIdx.x & 31;
  const int wave = threadIdx.x >> 5;
  const int half = lane >> 4;        // 0: lanes 0-15, 1: lanes 16-31
  const int mr   = lane & 15;        // row (A) / col (B,D) within tile
  const int col0 = wave << 4;

  v8f acc = {0.f, 0.f, 0.f, 0.f, 0.f, 0.f, 0.f, 0.f};
  const float* sArow = &sA[mr * kLdsStride];

  for (int k0 = 0; k0 < K; k0 += 4) {
    const int ka = k0 + (half << 1);
    v2f a; a.x = sArow[ka]; a.y = sArow[ka + 1];
    v2f b; b.x = W[ka * kHid + col0 + mr];
    b.y        = W[(ka + 1) * kHid + col0 + mr];
    acc = __builtin_amdgcn_wmma_f32_16x16x4_f32(
        false, a, false, b, (short)0, acc, false, false);
  }

  float* out = C + (size_t)(row0 + (half << 3)) * kHid + col0 + mr;
#pragma unroll
  for (int r = 0; r < 8; ++r) out[(size_t)r * kHid] = acc[r];
}

// ---------------- fused GCN aggregation (gather form, no f32 atomics) ----------
// out[d] = relu( dinv[d]*( sum_{s in adj(d)} dinv[s]*h[s] + dinv[d]*h[d] ) + bias )
// One wave per node; lane handles 4 channels (float4). Next-row software prefetch.
__global__ __launch_bounds__(256) void k_gather(
    const float* __restrict__ h, const int* __restrict__ rowptr,
    const int* __restrict__ adj, const float* __restrict__ dinv,
    const float* __restrict__ bias, float* __restrict__ out) {
  int node = (blockIdx.x << 3) + (threadIdx.x >> 5);
  if (node >= kNodes) return;
  const int lane = threadIdx.x & 31;
  const int c = lane << 2;

  const int beg = rowptr[node];
  const int end = rowptr[node + 1];
  const float dd = dinv[node];

  float4 selfv = *reinterpret_cast<const float4*>(h + (size_t)node * kHid + c);
  float4 acc;
  acc.x = selfv.x * dd; acc.y = selfv.y * dd;
  acc.z = selfv.z * dd; acc.w = selfv.w * dd;

  int s = (beg < end) ? adj[beg] : 0;
  for (int i = beg; i < end; ++i) {
    int snext = (i + 1 < end) ? adj[i + 1] : s;
    __builtin_prefetch(h + (size_t)snext * kHid + c, 0, 1);  // global_prefetch_b8
    float w = dinv[s];
    float4 v = *reinterpret_cast<const float4*>(h + (size_t)s * kHid + c);
    acc.x += v.x * w; acc.y += v.y * w;
    acc.z += v.z * w; acc.w += v.w * w;
    s = snext;
  }

  float4 o;
  o.x = fmaxf(acc.x * dd + bias[c + 0], 0.f);
  o.y = fmaxf(acc.y * dd + bias[c + 1], 0.f);
  o.z = fmaxf(acc.z * dd + bias[c + 2], 0.f);
  o.w = fmaxf(acc.w * dd + bias[c + 3], 0.f);
  *reinterpret_cast<float4*>(out + (size_t)node * kHid + c) = o;
}

// ---------------- global mean pool (batch is sorted: run-accumulate) ----------
// One wave per run of 64 consecutive nodes; flush atomics only on graph change.
__global__ __launch_bounds__(256) void k_pool_sum(
    const float* __restrict__ h, const int* __restrict__ batch,
    float* __restrict__ pooled, float* __restrict__ cnt) {
  int chunk = (blockIdx.x << 3) + (threadIdx.x >> 5);
  int n0 = chunk << 6;
  if (n0 >= kNodes) return;
  int n1 = n0 + 64; if (n1 > kNodes) n1 = kNodes;
  const int lane = threadIdx.x & 31;
  const int c = lane << 2;

  int gcur = batch[n0];
  float4 acc = make_float4(0.f, 0.f, 0.f, 0.f);
  float count = 0.f;
  for (int n = n0; n < n1; ++n) {
    int g = batch[n];
    if (g != gcur) {
      float* p = pooled + gcur * kHid + c;
      atomAddF(p + 0, acc.x); atomAddF(p + 1, acc.y);
      atomAddF(p + 2, acc.z); atomAddF(p + 3, acc.w);
      if (lane == 0) atomAddF(cnt + gcur, count);
      acc = make_float4(0.f, 0.f, 0.f, 0.f);
      count = 0.f;
      gcur = g;
    }
    float4 v = *reinterpret_cast<const float4*>(h + (size_t)n * kHid + c);
    acc.x += v.x; acc.y += v.y; acc.z += v.z; acc.w += v.w;
    count += 1.f;
  }
  float* p = pooled + gcur * kHid + c;
  atomAddF(p + 0, acc.x); atomAddF(p + 1, acc.y);
  atomAddF(p + 2, acc.z); atomAddF(p + 3, acc.w);
  if (lane == 0) atomAddF(cnt + gcur, count);
}

__global__ void k_pool_div(float* __restrict__ pooled, const float* __restrict__ cnt) {
  int i = blockIdx.x * blockDim.x + threadIdx.x;      // kGraphs*kHid
  if (i < kGraphs * kHid) pooled[i] /= fmaxf(cnt[i >> 7], 1.0f);
}

// ---------------- classifier head (trivial FLOPs, scalar) ----------------
__global__ void k_mlp1(const float* __restrict__ pooled, const float* __restrict__ Wl1,
                       const float* __restrict__ bl1, float* __restrict__ hidden) {
  int idx = blockIdx.x * blockDim.x + threadIdx.x;    // 64*64
  if (idx >= kGraphs * kHalfHid) return;
  int g = idx >> 6, j = idx & 63;
  float s = bl1[j];
  const float* pr = pooled + g * kHid;
#pragma unroll 8
  for (int k = 0; k < kHid; ++k) s += pr[k] * Wl1[k * kHalfHid + j];
  hidden[idx] = fmaxf(s, 0.f);
}

__global__ void k_mlp2(const float* __restrict__ hidden, const float* __restrict__ Wl2,
                       const float* __restrict__ bl2, float* __restrict__ out) {
  int idx = blockIdx.x * blockDim.x + threadIdx.x;    // 64*10
  if (idx >= kGraphs * kNcls) return;
  int g = idx / kNcls, j = idx - g * kNcls;
  float s = bl2[j];
  const float* hr = hidden + g * kHalfHid;
#pragma unroll 8
  for (int k = 0; k < kHalfHid; ++k) s += hr[k] * Wl2[k * kNcls + j];
  out[idx] = s;
}

// ---------------- launch ----------------
extern "C" void kernel_launch(void* const* d_in, const int* in_sizes, int n_in,
                              void* d_out, int out_size, void* d_ws, size_t ws_size,
                              hipStream_t stream) {
  const float* x   = (const float*)d_in[0];
  const int*   src = (const int*)d_in[1];
  const int*   dst = (const int*)d_in[2];
  const int*   bat = (const int*)d_in[3];
  const float* W1  = (const float*)d_in[4];
  const float* b1  = (const float*)d_in[5];
  const float* W2  = (const float*)d_in[6];
  const float* b2  = (const float*)d_in[7];
  const float* Wl1 = (const float*)d_in[8];
  const float* bl1 = (const float*)d_in[9];
  const float* Wl2 = (const float*)d_in[10];
  const float* bl2 = (const float*)d_in[11];
  float* out = (float*)d_out;

  // Workspace: dinv | bufA(N*H) | bufB(N*H) | pooled | cntf | hidden | ints...
  float* ws      = (float*)d_ws;
  float* dinv    = ws;
  float* bufA    = dinv + kNodes;
  float* bufB    = bufA + (size_t)kNodes * kHid;
  float* pooled  = bufB + (size_t)kNodes * kHid;
  float* cntf    = pooled + kGraphs * kHid;
  float* hidden  = cntf + kGraphs;
  int*   cnt     = (int*)(hidden + kGraphs * kHalfHid);
  int*   rowptr  = cnt + kNodes;              // kNodes+1 entries
  int*   rowwork = rowptr + kNodes + 1;
  int*   adj     = rowwork + kNodes;          // kEdges entries
  int*   partial = adj + kEdges;              // kScanBlks entries

  const dim3 blk(256);
  const int nodeBlks = (kNodes + 255) / 256;  // 391
  const int edgeBlks = kEdges / 256;          // 6250
  const int gemmBlks = kNodes / 16;           // 6250
  const int gathBlks = kNodes / 8;            // 12500 (wave per node)
  const int poolBlks = ((kNodes + 63) / 64 + 7) / 8;   // wave per 64-node run

  // ---- CSR build (once per call; reused by both layers) ----
  k_zero_i<<<nodeBlks, blk, 0, stream>>>(cnt, kNodes);
  k_count<<<edgeBlks, blk, 0, stream>>>(dst, cnt);
  k_dinv<<<nodeBlks, blk, 0, stream>>>(cnt, dinv);
  k_scan_p1<<<kScanBlks, blk, 0, stream>>>(cnt, rowptr, partial);
  k_scan_p2<<<1, 512, 0, stream>>>(partial);
  k_scan_p3<<<kScanBlks, blk, 0, stream>>>(rowptr, partial);
  (void)hipMemcpyAsync(rowwork, rowptr, sizeof(int) * kNodes,
                       hipMemcpyDeviceToDevice, stream);
  k_fill_adj<<<edgeBlks, blk, 0, stream>>>(src, dst, rowwork, adj);

  // ---- layer 1: h = x@W1 ; fused gather+selfloop+bias+relu ----
  k_gemm_wmma<<<gemmBlks, blk, 0, stream>>>(x, W1, bufA, kInDim);
  k_gather<<<gathBlks, blk, 0, stream>>>(bufA, rowptr, adj, dinv, b1, bufB);

  // ---- layer 2 ----
  k_gemm_wmma<<<gemmBlks, blk, 0, stream>>>(bufB, W2, bufA, kHid);
  k_gather<<<gathBlks, blk, 0, stream>>>(bufA, rowptr, adj, dinv, b2, bufB);

  // ---- mean pool per graph ----
  k_zero_f<<<(kGraphs * kHid + kGraphs + 255) / 256, blk, 0, stream>>>(
      pooled, kGraphs * kHid + kGraphs);
  k_pool_sum<<<poolBlks, blk, 0, stream>>>(bufB, bat, pooled, cntf);
  k_pool_div<<<(kGraphs * kHid + 255) / 256, blk, 0, stream>>>(pooled, cntf);

  // ---- classifier head ----
  k_mlp1<<<(kGraphs * kHalfHid + 255) / 256, blk, 0, stream>>>(pooled, Wl1, bl1, hidden);
  k_mlp2<<<(kGraphs * kNcls + 255) / 256, blk, 0, stream>>>(hidden, Wl2, bl2, out);
}